// GNN_Model_14654428414305
// MI455X (gfx1250) — compile-verified
//
#include <hip/hip_runtime.h>
#include <hip/hip_bf16.h>

// GAT (3 layers, heads 4/4/1, C=64) + final FC. fp32 throughout, matching the
// reference. GEMMs use V_WMMA_F32_16X16X4_F32. Edge softmax/aggregation is
// scatter-based with fp32 atomics (L2-resident: h is 51MB < 192MB L2).
//
// Workspace layout (floats): bufA[N*256] | bufB[N*256] | es[N*4] | ed[N*4] |
// m[N*4 u32] | s[N*4] | p[(E+N)*4]  => ~119.2 MB total.

#define NNODES 50000
#define NEDGES 800000
#define INDIM 128
#define CHN 64
#define NCLASS 10

typedef __attribute__((ext_vector_type(2))) float v2f;
typedef __attribute__((ext_vector_type(8))) float v8f;

// ---------------------------------------------------------------- GEMM (WMMA)
// C[M,N] = A[M,K] @ B[K,N].  Requires K%16==0, N%64==0.  Block = 256 thr =
// 8 waves; block tile 64x64; each wave owns one 16-row stripe and two 16-col
// tiles. K staged through LDS in blocks of 16 (4 wmma k-steps).
__global__ __launch_bounds__(256) void gemm_wmma_f32(
    const float* __restrict__ A, const float* __restrict__ B,
    float* __restrict__ C, int M, int K, int N) {
  __shared__ float As[64][17];   // +1 pad: conflict-free column reads
  __shared__ float Bs[16][64];

  const int tid  = threadIdx.x;
  const int wave = tid >> 5;
  const int lane = tid & 31;
  const int half = lane >> 4;    // 0: K pair {0,1}; 1: K pair {2,3}
  const int l    = lane & 15;

  const int rowBase = blockIdx.y * 64;
  const int colBase = blockIdx.x * 64;
  const int wm  = wave >> 1;         // 0..3 : 16-row stripe in block tile
  const int wnb = (wave & 1) * 2;    // 0 or 2 : first of two 16-col tiles

  v8f acc0 = {};
  v8f acc1 = {};

  const int arow = tid >> 2;         // 0..63
  const int akc  = (tid & 3) << 2;   // 0,4,8,12
  const int brow = tid >> 4;         // 0..15
  const int bcol = (tid & 15) << 2;  // 0..60

  for (int k0 = 0; k0 < K; k0 += 16) {
    // stage A 64x16 (row-guarded) and B 16x64
    {
      int gr = rowBase + arow;
      float4 av = make_float4(0.f, 0.f, 0.f, 0.f);
      if (gr < M) av = *(const float4*)(A + (size_t)gr * K + k0 + akc);
      As[arow][akc + 0] = av.x; As[arow][akc + 1] = av.y;
      As[arow][akc + 2] = av.z; As[arow][akc + 3] = av.w;
    }
    {
      float4 bv = *(const float4*)(B + (size_t)(k0 + brow) * N + colBase + bcol);
      *(float4*)(&Bs[brow][bcol]) = bv;
    }
    __syncthreads();

#pragma unroll
    for (int kk = 0; kk < 16; kk += 4) {
      const int ka = kk + (half << 1);
      v2f a;
      a.x = As[wm * 16 + l][ka];
      a.y = As[wm * 16 + l][ka + 1];
      v2f b0, b1;
      b0.x = Bs[ka][wnb * 16 + l];
      b0.y = Bs[ka + 1][wnb * 16 + l];
      b1.x = Bs[ka][(wnb + 1) * 16 + l];
      b1.y = Bs[ka + 1][(wnb + 1) * 16 + l];
      acc0 = __builtin_amdgcn_wmma_f32_16x16x4_f32(false, a, false, b0,
                                                   (short)0, acc0, false, false);
      acc1 = __builtin_amdgcn_wmma_f32_16x16x4_f32(false, a, false, b1,
                                                   (short)0, acc1, false, false);
    }
    __syncthreads();
  }

#pragma unroll
  for (int i = 0; i < 8; ++i) {
    int gr = rowBase + wm * 16 + i + half * 8;
    if (gr < M) {
      C[(size_t)gr * N + colBase + wnb * 16 + l]       = acc0[i];
      C[(size_t)gr * N + colBase + (wnb + 1) * 16 + l] = acc1[i];
    }
  }
}

// ------------------------------------------------------------- misc kernels
__global__ void zero_f32(float* __restrict__ p, int n) {
  int i = blockIdx.x * blockDim.x + threadIdx.x;
  if (i < n) p[i] = 0.0f;
}

// e_src/e_dst per (node, head): dot(h[n,h,:], a[h,:])
__global__ void node_scores(const float* __restrict__ h,
                            const float* __restrict__ a_src,
                            const float* __restrict__ a_dst,
                            float* __restrict__ es, float* __restrict__ ed,
                            int H) {
  int idx = blockIdx.x * blockDim.x + threadIdx.x;
  if (idx >= NNODES * H) return;
  int n = idx / H, hh = idx - n * H;
  const float* hp = h + (size_t)n * H * CHN + hh * CHN;
  const float* as = a_src + hh * CHN;
  const float* ad = a_dst + hh * CHN;
  float ss = 0.f, dd = 0.f;
#pragma unroll 8
  for (int c = 0; c < CHN; ++c) {
    float v = hp[c];
    ss += v * as[c];
    dd += v * ad[c];
  }
  es[idx] = ss;
  ed[idx] = dd;
}

// order-preserving float<->uint map for atomic max (exact float ordering)
__device__ __forceinline__ unsigned ordmap(float f) {
  unsigned u = __float_as_uint(f);
  return (u & 0x80000000u) ? ~u : (u | 0x80000000u);
}
__device__ __forceinline__ float ordunmap(unsigned u) {
  return __uint_as_float((u & 0x80000000u) ? (u & 0x7FFFFFFFu) : ~u);
}

__device__ __forceinline__ void edge_ends(const int* __restrict__ ei, int e,
                                          int& s, int& d) {
  if (e < NEDGES) {
    s = ei[e];
    d = ei[NEDGES + e];
  } else {
    s = d = e - NEDGES;  // self loop
  }
}

// per (edge, head): leaky_relu logit; scatter-max into m
__global__ void edge_logits(const float* __restrict__ es,
                            const float* __restrict__ ed,
                            const int* __restrict__ ei,
                            float* __restrict__ ebuf,
                            unsigned* __restrict__ m, int H) {
  int idx = blockIdx.x * blockDim.x + threadIdx.x;
  int ET = NEDGES + NNODES;
  if (idx >= ET * H) return;
  int e = idx / H, hh = idx - e * H;
  int s, d;
  edge_ends(ei, e, s, d);
  float v = es[s * H + hh] + ed[d * H + hh];
  v = (v > 0.f) ? v : 0.2f * v;  // leaky_relu, slope 0.2
  ebuf[idx] = v;
  atomicMax(&m[d * H + hh], ordmap(v));
}

// per (edge, head): p = exp(e - m[dst]); scatter-sum into s
__global__ void edge_softmax(const int* __restrict__ ei,
                             float* __restrict__ ebuf,
                             const unsigned* __restrict__ m,
                             float* __restrict__ sbuf, int H) {
  int idx = blockIdx.x * blockDim.x + threadIdx.x;
  int ET = NEDGES + NNODES;
  if (idx >= ET * H) return;
  int e = idx / H, hh = idx - e * H;
  int s, d;
  edge_ends(ei, e, s, d);
  (void)s;
  float p = expf(ebuf[idx] - ordunmap(m[d * H + hh]));
  ebuf[idx] = p;
  atomicAdd(&sbuf[d * H + hh], p);
}

// per (edge, feature): out[dst] += p * h[src]  (unnormalized; divide later)
__global__ void edge_aggregate(const int* __restrict__ ei,
                               const float* __restrict__ pbuf,
                               const float* __restrict__ h,
                               float* __restrict__ out, int H, int hcShift) {
  int idx = blockIdx.x * blockDim.x + threadIdx.x;
  int HC = H * CHN;
  int ET = NEDGES + NNODES;
  if (idx >= ET * HC) return;
  int e = idx >> hcShift;
  int f = idx & (HC - 1);
  int s, d;
  edge_ends(ei, e, s, d);
  int hh = f >> 6;  // CHN == 64
  float p = pbuf[e * H + hh];
  atomicAdd(&out[(size_t)d * HC + f], p * h[(size_t)s * HC + f]);
}

// per (node, feature): out = out / (s + 1e-16) + bias ; ReLU
__global__ void node_finalize(float* __restrict__ out,
                              const float* __restrict__ sbuf,
                              const float* __restrict__ bias, int H,
                              int hcShift) {
  int idx = blockIdx.x * blockDim.x + threadIdx.x;
  int HC = H * CHN;
  if (idx >= NNODES * HC) return;
  int n = idx >> hcShift;
  int f = idx & (HC - 1);
  int hh = f >> 6;
  float v = out[idx] / (sbuf[n * H + hh] + 1e-16f) + bias[f];
  out[idx] = (v > 0.f) ? v : 0.f;
}

// final FC: [N,64] @ [64,10] + b
__global__ void fc_kernel(const float* __restrict__ h,
                          const float* __restrict__ W,
                          const float* __restrict__ b,
                          float* __restrict__ out) {
  int idx = blockIdx.x * blockDim.x + threadIdx.x;
  if (idx >= NNODES * NCLASS) return;
  int n = idx / NCLASS, j = idx - n * NCLASS;
  float acc = b[j];
#pragma unroll
  for (int c = 0; c < CHN; ++c) acc += h[(size_t)n * CHN + c] * W[c * NCLASS + j];
  out[idx] = acc;
}

// ------------------------------------------------------------------- driver
extern "C" void kernel_launch(void* const* d_in, const int* in_sizes, int n_in,
                              void* d_out, int out_size, void* d_ws,
                              size_t ws_size, hipStream_t stream) {
  const float* x    = (const float*)d_in[0];
  const int*   ei   = (const int*)d_in[1];
  const float* W1   = (const float*)d_in[2];
  const float* a1s  = (const float*)d_in[3];
  const float* a1d  = (const float*)d_in[4];
  const float* b1   = (const float*)d_in[5];
  const float* W2   = (const float*)d_in[6];
  const float* a2s  = (const float*)d_in[7];
  const float* a2d  = (const float*)d_in[8];
  const float* b2   = (const float*)d_in[9];
  const float* W3   = (const float*)d_in[10];
  const float* a3s  = (const float*)d_in[11];
  const float* a3d  = (const float*)d_in[12];
  const float* b3   = (const float*)d_in[13];
  const float* fcW  = (const float*)d_in[14];
  const float* fcb  = (const float*)d_in[15];

  float* ws   = (float*)d_ws;
  float* bufA = ws;                                   // N*256
  float* bufB = bufA + (size_t)NNODES * 256;          // N*256
  float* esb  = bufB + (size_t)NNODES * 256;          // N*4
  float* edb  = esb + (size_t)NNODES * 4;             // N*4
  unsigned* mb = (unsigned*)(edb + (size_t)NNODES * 4);  // N*4
  float* sb   = (float*)mb + (size_t)NNODES * 4;      // N*4
  float* pbuf = sb + (size_t)NNODES * 4;              // (E+N)*4

  const int ET = NEDGES + NNODES;

  auto layer = [&](const float* in, int Kin, int H, const float* W,
                   const float* as, const float* ad, const float* bias,
                   float* hbuf, float* obuf) {
    const int HC = H * CHN;
    const int hcShift = (HC == 256) ? 8 : 6;
    dim3 gg(HC / 64, (NNODES + 63) / 64);
    gemm_wmma_f32<<<gg, 256, 0, stream>>>(in, W, hbuf, NNODES, Kin, HC);
    node_scores<<<(NNODES * H + 255) / 256, 256, 0, stream>>>(hbuf, as, ad,
                                                              esb, edb, H);
    // zero m (ordmap sentinel 0 < any real float), s, and out accumulator
    zero_f32<<<(NNODES * H + 255) / 256, 256, 0, stream>>>((float*)mb,
                                                           NNODES * H);
    zero_f32<<<(NNODES * H + 255) / 256, 256, 0, stream>>>(sb, NNODES * H);
    zero_f32<<<(NNODES * HC + 255) / 256, 256, 0, stream>>>(obuf, NNODES * HC);
    edge_logits<<<(ET * H + 255) / 256, 256, 0, stream>>>(esb, edb, ei, pbuf,
                                                          mb, H);
    edge_softmax<<<(ET * H + 255) / 256, 256, 0, stream>>>(ei, pbuf, mb, sb, H);
    edge_aggregate<<<((long long)ET * HC + 255) / 256, 256, 0, stream>>>(
        ei, pbuf, hbuf, obuf, H, hcShift);
    node_finalize<<<(NNODES * HC + 255) / 256, 256, 0, stream>>>(obuf, sb,
                                                                 bias, H,
                                                                 hcShift);
  };

  // L1: x[N,128] -> h in bufA, out (relu'd) in bufB
  layer(x, INDIM, 4, W1, a1s, a1d, b1, bufA, bufB);
  // L2: bufB -> h in bufA, out in bufB (gemm reads bufB before zero overwrites)
  layer(bufB, 256, 4, W2, a2s, a2d, b2, bufA, bufB);
  // L3: bufB -> h in bufA (N*64), out in bufB (N*64)
  layer(bufB, 256, 1, W3, a3s, a3d, b3, bufA, bufB);
  // FC
  fc_kernel<<<(NNODES * NCLASS + 255) / 256, 256, 0, stream>>>(bufB, fcW, fcb,
                                                               (float*)d_out);
}